// CarotidArtery_Centerline_Graph_59596966199698
// MI455X (gfx1250) — compile-verified
//
#include <hip/hip_runtime.h>
#include <stdint.h>

#define BATCH 4
#define NPTS  32768
#define FCL   2048
#define DCH   256
#define NT    32
#define NTHR  256
#define MIN_W 1e-10f

typedef unsigned int u32x4 __attribute__((ext_vector_type(4)));
typedef int          i32x8 __attribute__((ext_vector_type(8)));
typedef int          i32x4 __attribute__((ext_vector_type(4)));

// out[b, c, n]:
//   c in [0,3): pv[b, n, c]                  (coords)
//   c == 3 (if has_extra): pv[b, n, 9]       (extra scalar)
//   c >= coff: sum_k 1/max(w_k,MIN_W) * feat[b, idx_k, c-coff]
__global__ __launch_bounds__(NTHR)
void interp_transpose_kernel(const float* __restrict__ pv,
                             const float* __restrict__ feat,
                             float* __restrict__ out,
                             int C, int has_extra) {
  extern __shared__ float smem[];
  float* pv_s  = smem;                                // NT*10 = 320 floats
  float* trans = smem + 320;                          // DCH*(NT+1) floats (padded vs bank conflicts)
  int*   gidx  = (int*)(smem + 320 + DCH * (NT + 1)); // NT*3 ints (pre-scaled row offsets)
  float* ginvw = (float*)(gidx + NT * 3);             // NT*3 floats

  const int t  = threadIdx.x;
  const int b  = blockIdx.y;
  const int n0 = blockIdx.x * NT;

  const float* pv_tile = pv + ((size_t)b * NPTS + n0) * 10;

  // ---- Stage pv tile (NT x 10 f32 = 1280 contiguous bytes) into LDS via the
  //      Tensor Data Mover: one async-tensor instruction instead of 10 loads.
  //      Descriptor per CDNA5 ISA ch.8 (1-D tile, data_size=4B, 320 elements).
  if (t < 32) {  // wave 0 only issues the TDM op (per-wave instruction)
    const uint64_t ga = (uint64_t)(uintptr_t)pv_tile;
    const uint32_t lds_base = (uint32_t)__builtin_amdgcn_groupstaticsize(); // dyn LDS start

    u32x4 g0;
    g0.x = 1u;                                                  // count = 1 (valid D#)
    g0.y = lds_base;                                            // lds_addr (bytes)
    g0.z = (uint32_t)ga;                                        // global_addr[31:0]
    g0.w = (uint32_t)((ga >> 32) & 0x01FFFFFFu) | (2u << 30);   // addr[56:32] | type=2

    const uint32_t nelem = (uint32_t)(NT * 10);                 // 320 x 4-byte elements
    i32x8 g1;
    g1[0] = (int)(2u << 16);                                    // data_size = 2 -> 4 bytes
    g1[1] = (int)((nelem & 0xFFFFu) << 16);                     // tensor_dim0[15:0] @ [63:48]
    g1[2] = (int)((nelem >> 16) | (1u << 16));                  // tensor_dim0[31:16] | tensor_dim1=1
    g1[3] = (int)(nelem << 16);                                 // tile_dim0 @ [127:112]
    g1[4] = 1;                                                  // tile_dim1 = 1, tile_dim2 = 0
    g1[5] = (int)nelem;                                         // tensor_dim0_stride[31:0]
    g1[6] = 0;
    g1[7] = 0;

    const i32x4 gz4 = {0, 0, 0, 0};                             // D# groups 2/3 (unused, 1-D tile)
    const i32x8 gz8 = {0, 0, 0, 0, 0, 0, 0, 0};                 // trailing descriptor words (unused)
    __builtin_amdgcn_tensor_load_to_lds(g0, g1, gz4, gz4, gz8, 0);
    __builtin_amdgcn_s_wait_tensorcnt(0);
  }
  __syncthreads();

  // ---- Precompute gather offsets + inverse weights once per block (shared) ----
  if (t < NT * 3) {
    const int n = t / 3, k = t % 3;
    int idx = (int)pv_s[n * 10 + 3 + k];          // trunc-toward-zero like astype(int32)
    idx = idx < 0 ? 0 : (idx > (FCL - 1) ? (FCL - 1) : idx);
    gidx[t]  = idx * DCH;
    ginvw[t] = 1.0f / fmaxf(pv_s[n * 10 + 6 + k], MIN_W);
  }
  __syncthreads();

  // ---- Gather + weighted sum: thread t owns channel d=t for all NT points.
  //      Consecutive lanes read consecutive bytes -> fully coalesced L2-hit loads.
  {
    const float* fb = feat + (size_t)b * FCL * DCH + t;
    float* trow = trans;  // indexed trans[d*(NT+1)+n], d = t
    #pragma unroll 4
    for (int n = 0; n < NT; ++n) {
      float acc = ginvw[n * 3 + 0] * fb[gidx[n * 3 + 0]];
      acc      += ginvw[n * 3 + 1] * fb[gidx[n * 3 + 1]];
      acc      += ginvw[n * 3 + 2] * fb[gidx[n * 3 + 2]];
      trow[t * (NT + 1) + n] = acc;               // stride-33 write: bank-conflict-free
    }
  }
  __syncthreads();

  // ---- Write transposed output: lanes cover consecutive n -> 128B/wave stores ----
  float* ob = out + (size_t)b * C * NPTS + n0;
  if (t < 96) {                                   // coord channels 0..2
    const int c = t >> 5, n = t & 31;
    ob[(size_t)c * NPTS + n] = pv_s[n * 10 + c];
  } else if (has_extra && t < 128) {              // extra channel 3
    const int n = t & 31;
    ob[(size_t)3 * NPTS + n] = pv_s[n * 10 + 9];
  }
  const int n     = t & 31;
  const int dbase = t >> 5;                       // 0..7
  const int coff  = 3 + has_extra;
  #pragma unroll
  for (int i = 0; i < 32; ++i) {
    const int d = dbase + 8 * i;
    ob[(size_t)(coff + d) * NPTS + n] = trans[d * (NT + 1) + n];
  }
}

extern "C" void kernel_launch(void* const* d_in, const int* in_sizes, int n_in,
                              void* d_out, int out_size, void* d_ws, size_t ws_size,
                              hipStream_t stream) {
  (void)in_sizes; (void)n_in; (void)d_ws; (void)ws_size; (void)out_size;
  const float* pv_o = (const float*)d_in[0];
  const float* pv_b = (const float*)d_in[1];
  const float* fo   = (const float*)d_in[2];
  const float* fb   = (const float*)d_in[3];

  float* out  = (float*)d_out;
  float* outo = out;                                          // [B, 259, N]
  float* outb = out + (size_t)BATCH * (3 + DCH) * NPTS;       // [B, 260, N]

  const size_t shmem = (320 + DCH * (NT + 1) + NT * 3 * 2) * sizeof(float);
  dim3 grid(NPTS / NT, BATCH);
  interp_transpose_kernel<<<grid, NTHR, shmem, stream>>>(pv_o, fo, outo, 3 + DCH, 0);
  interp_transpose_kernel<<<grid, NTHR, shmem, stream>>>(pv_b, fb, outb, 4 + DCH, 1);
}